// L5_77206332113745
// MI455X (gfx1250) — compile-verified
//
#include <hip/hip_runtime.h>
#include <hip/hip_bf16.h>
#include <math.h>

typedef __attribute__((ext_vector_type(2))) float v2f;
typedef __attribute__((ext_vector_type(8))) float v8f;

#if __has_builtin(__builtin_amdgcn_global_load_async_to_lds_b128) && \
    __has_builtin(__builtin_amdgcn_s_wait_asynccnt)
#define USE_ASYNC_LDS 1
#else
#define USE_ASYNC_LDS 0
#warning "async-to-LDS builtins unavailable; falling back to ds_store staging"
#endif

// Builtin signature (from hipcc diagnostic): param0 is
// 'int __vector(4) __device__ *' i.e. v4i in addrspace(1); param1 is the LDS side.
typedef int v4i_mem __attribute__((vector_size(16)));
typedef __attribute__((address_space(1))) v4i_mem* gptr_v4i;
typedef __attribute__((address_space(3))) v4i_mem* lptr_v4i;

__device__ __forceinline__ void cp16_to_lds(float* lp, const float* gp) {
#if USE_ASYNC_LDS
  __builtin_amdgcn_global_load_async_to_lds_b128(
      (gptr_v4i)(uintptr_t)gp, (lptr_v4i)(uint32_t)(uintptr_t)lp, 0, 0);
#else
  *(float4*)lp = *(const float4*)gp;
#endif
}

__device__ __forceinline__ float elu_f(float x) { return x > 0.f ? x : expm1f(x); }

// ---------------- concat(one_hot, features) -> [N,64] ----------------
__global__ void concat_kernel(const float* __restrict__ oh, const float* __restrict__ ft,
                              float* __restrict__ xc, int n) {
  int idx = blockIdx.x * blockDim.x + threadIdx.x;
  if (idx >= n * 64) return;
  int i = idx >> 6, j = idx & 63;
  xc[idx] = (j < 20) ? oh[i * 20 + j] : ft[i * 44 + (j - 20)];
}

// ---------------- y = elu(x @ W + b), thread per (row, out) ----------------
__global__ void linear_elu_kernel(const float* __restrict__ x, const float* __restrict__ W,
                                  const float* __restrict__ b, float* __restrict__ y,
                                  int n, int in, int out) {
  int idx = blockIdx.x * blockDim.x + threadIdx.x;
  if (idx >= n * out) return;
  int i = idx / out, o = idx - i * out;
  float s = b[o];
  for (int k = 0; k < in; ++k) s = fmaf(x[(size_t)i * in + k], W[k * out + o], s);
  y[idx] = elu_f(s);
}

// ------- hT[o][i] = sum_k x[i][k] * W[k][o]; rows o>=out zero-padded -------
__global__ void hlinT_kernel(const float* __restrict__ x, const float* __restrict__ W,
                             float* __restrict__ hT, int n, int in, int out) {
  int i = blockIdx.x * blockDim.x + threadIdx.x;
  int o = blockIdx.y;
  if (i >= n) return;
  float s = 0.f;
  if (o < out) {
    for (int k = 0; k < in; ++k) s = fmaf(x[(size_t)i * in + k], W[k * out + o], s);
  }
  hT[(size_t)o * n + i] = s;
}

// ---------------- partial y = A[:, seg] @ h[seg, :] via WMMA, LDS-tiled ----------------
// Block: 256 threads = 8 waves; each wave owns one 16-row strip (128 rows/block).
// gridDim.y = K segments. A tile (128 x KC) + shared hT tile (NT*16 x KC) are staged
// into double-buffered LDS with async-to-LDS copies; fragments come from LDS (ds_load_b64).
template <int NT>
__launch_bounds__(256)
__global__ void spmm_wmma_kernel(const float* __restrict__ A,
                                 const float* __restrict__ hT,
                                 float* __restrict__ yseg,
                                 int n, int out, int segLen) {
  constexpr int KC = 64;            // K per chunk
  constexpr int RS = KC + 4;        // LDS row stride (dwords): banks 4l+j, conflict-free
  constexpr int AROWS = 128, BROWS = NT * 16;
  constexpr int ASZ = AROWS * RS, BSZ = BROWS * RS;
  constexpr int AUNITS = AROWS * (KC / 4);   // 16B units per A tile
  constexpr int BUNITS = BROWS * (KC / 4);
  constexpr int NSTAGE = AUNITS / 256 + (BUNITS + 255) / 256;  // async ops/thread/stage

  extern __shared__ float lds[];  // 2 * (ASZ + BSZ) floats

  const int tid   = threadIdx.x;
  const int lane  = tid & 31;
  const int wave  = tid >> 5;
  const int l15   = lane & 15;
  const int koff  = (lane >> 4) << 1;  // 0 or 2
  const int m0blk = blockIdx.x * AROWS;
  const int seg   = blockIdx.y;
  const int k0seg = seg * segLen;
  const int nchunks = segLen / KC;

  auto stage = [&](int buf, int k0) {
    float* LA = lds + buf * (ASZ + BSZ);
    float* LB = LA + ASZ;
    for (int u = tid; u < AUNITS; u += 256) {
      int row = u >> 4, c4 = (u & 15) << 2;
      cp16_to_lds(&LA[row * RS + c4], A + (size_t)(m0blk + row) * n + k0 + c4);
    }
    for (int u = tid; u < BUNITS; u += 256) {
      int row = u >> 4, c4 = (u & 15) << 2;
      cp16_to_lds(&LB[row * RS + c4], hT + (size_t)row * n + k0 + c4);
    }
  };

  v8f acc[NT] = {};
  stage(0, k0seg);

  for (int c = 0; c < nchunks; ++c) {
    const int buf = c & 1;
    if (c + 1 < nchunks) {
      stage(buf ^ 1, k0seg + (c + 1) * KC);   // prefetch next chunk (async DMA)
#if USE_ASYNC_LDS
      __builtin_amdgcn_s_wait_asynccnt(NSTAGE);  // only chunk c's copies must be done
#endif
    } else {
#if USE_ASYNC_LDS
      __builtin_amdgcn_s_wait_asynccnt(0);
#endif
    }
    __syncthreads();

    const float* La = lds + buf * (ASZ + BSZ) + (wave * 16 + l15) * RS + koff;
    const float* Lb = lds + buf * (ASZ + BSZ) + ASZ + l15 * RS + koff;
#if __has_builtin(__builtin_amdgcn_wmma_f32_16x16x4_f32)
#pragma unroll 4
    for (int ks = 0; ks < KC; ks += 4) {
      v2f a = *(const v2f*)(La + ks);
#pragma unroll
      for (int t = 0; t < NT; ++t) {
        v2f b = *(const v2f*)(Lb + t * 16 * RS + ks);
        acc[t] = __builtin_amdgcn_wmma_f32_16x16x4_f32(false, a, false, b,
                                                       (short)0, acc[t], false, false);
      }
    }
#endif
    __syncthreads();  // protect buffer reused by chunk c+2's staging
  }

  // C/D layout: VGPR r -> row m0 + r + 8*(lane>=16); col = t*16 + (lane&15)
  const int rbase = m0blk + wave * 16 + ((lane >> 4) << 3);
  float* yp = yseg + (size_t)seg * n * out;
#pragma unroll
  for (int t = 0; t < NT; ++t) {
    int col = t * 16 + l15;
    if (col < out) {
#pragma unroll
      for (int r = 0; r < 8; ++r) {
        yp[(size_t)(rbase + r) * out + col] = acc[t][r];
      }
    }
  }
}

// ---------------- x_next = elu(sum_seg yseg + bias) ----------------
__global__ void reduce_bias_elu_kernel(const float* __restrict__ yseg,
                                       const float* __restrict__ bias,
                                       float* __restrict__ y, int n, int out, int S) {
  int idx = blockIdx.x * blockDim.x + threadIdx.x;
  if (idx >= n * out) return;
  int o = idx % out;
  float s = 0.f;
  for (int k = 0; k < S; ++k) s += yseg[(size_t)k * n * out + idx];
  s += bias[o];
  y[idx] = elu_f(s);
}

// ---------------- final MLP: 4 -> 8 -> 4 -> 1, sigmoid ----------------
__global__ void final_mlp_kernel(const float* __restrict__ x,
                                 const float* __restrict__ W1, const float* __restrict__ b1,
                                 const float* __restrict__ W2, const float* __restrict__ b2,
                                 const float* __restrict__ W3, const float* __restrict__ b3,
                                 float* __restrict__ out, int n) {
  int i = blockIdx.x * blockDim.x + threadIdx.x;
  if (i >= n) return;
  float x0[4];
#pragma unroll
  for (int k = 0; k < 4; ++k) x0[k] = x[(size_t)i * 4 + k];
  float h1[8];
#pragma unroll
  for (int o = 0; o < 8; ++o) {
    float s = b1[o];
#pragma unroll
    for (int k = 0; k < 4; ++k) s = fmaf(x0[k], W1[k * 8 + o], s);
    h1[o] = elu_f(s);
  }
  float h2[4];
#pragma unroll
  for (int o = 0; o < 4; ++o) {
    float s = b2[o];
#pragma unroll
    for (int k = 0; k < 8; ++k) s = fmaf(h1[k], W2[k * 4 + o], s);
    h2[o] = elu_f(s);
  }
  float s = b3[0];
#pragma unroll
  for (int k = 0; k < 4; ++k) s = fmaf(h2[k], W3[k], s);
  out[i] = 1.f / (1.f + expf(-s));
}

extern "C" void kernel_launch(void* const* d_in, const int* in_sizes, int n_in,
                              void* d_out, int out_size, void* d_ws, size_t ws_size,
                              hipStream_t stream) {
  const float* one_hot  = (const float*)d_in[0];
  const float* features = (const float*)d_in[1];
  // d_in[2] = gemme_features: unused by the reference
  const float* A   = (const float*)d_in[3];  // [1, N, N]
  const float* We1 = (const float*)d_in[4];  const float* be1 = (const float*)d_in[5];
  const float* We2 = (const float*)d_in[6];  const float* be2 = (const float*)d_in[7];
  const float* We3 = (const float*)d_in[8];  const float* be3 = (const float*)d_in[9];
  const float* Wg[5] = {(const float*)d_in[10], (const float*)d_in[12], (const float*)d_in[14],
                        (const float*)d_in[16], (const float*)d_in[18]};
  const float* bg[5] = {(const float*)d_in[11], (const float*)d_in[13], (const float*)d_in[15],
                        (const float*)d_in[17], (const float*)d_in[19]};
  const float* Wf1 = (const float*)d_in[20]; const float* bf1 = (const float*)d_in[21];
  const float* Wf2 = (const float*)d_in[22]; const float* bf2 = (const float*)d_in[23];
  const float* Wf3 = (const float*)d_in[24]; const float* bf3 = (const float*)d_in[25];

  const int n = in_sizes[0] / 20;  // N = 12288
  const int S = 4;                 // K-split segments
  const int segLen = n / S;

  float* ws    = (float*)d_ws;
  float* xbuf0 = ws;                      // n * 128
  float* xbuf1 = ws + (size_t)n * 128;    // n * 64
  float* hT    = ws + (size_t)n * 192;    // n * 64 (transposed, zero-padded feature map)
  float* yseg  = ws + (size_t)n * 256;    // S * n * 64 partial tiles

  const int T = 256;
  // --- encoder ---
  concat_kernel<<<(n * 64 + T - 1) / T, T, 0, stream>>>(one_hot, features, hT, n);
  linear_elu_kernel<<<(n * 32 + T - 1) / T, T, 0, stream>>>(hT, We1, be1, xbuf1, n, 64, 32);
  linear_elu_kernel<<<(n * 64 + T - 1) / T, T, 0, stream>>>(xbuf1, We2, be2, hT, n, 32, 64);
  linear_elu_kernel<<<(n * 128 + T - 1) / T, T, 0, stream>>>(hT, We3, be3, xbuf0, n, 64, 128);

  const dim3 sgrid(n / 128, S);
  constexpr int RS = 64 + 4;
  const size_t shA   = (size_t)128 * RS;
  auto shmem = [&](int NT) { return (unsigned)(2 * (shA + (size_t)NT * 16 * RS) * sizeof(float)); };

  struct Layer { int in, out, NT; const float *xin; float *xout; };
  const Layer L[5] = {
      {128, 64, 4, xbuf0, xbuf1},
      { 64, 32, 2, xbuf1, xbuf0},
      { 32, 16, 1, xbuf0, xbuf1},
      { 16,  8, 1, xbuf1, xbuf0},
      {  8,  4, 1, xbuf0, xbuf1},
  };
  for (int l = 0; l < 5; ++l) {
    const int outp = L[l].NT * 16;
    hlinT_kernel<<<dim3((n + T - 1) / T, outp), T, 0, stream>>>(L[l].xin, Wg[l], hT, n,
                                                                L[l].in, L[l].out);
    if (L[l].NT == 4)
      spmm_wmma_kernel<4><<<sgrid, 256, shmem(4), stream>>>(A, hT, yseg, n, L[l].out, segLen);
    else if (L[l].NT == 2)
      spmm_wmma_kernel<2><<<sgrid, 256, shmem(2), stream>>>(A, hT, yseg, n, L[l].out, segLen);
    else
      spmm_wmma_kernel<1><<<sgrid, 256, shmem(1), stream>>>(A, hT, yseg, n, L[l].out, segLen);
    reduce_bias_elu_kernel<<<(n * L[l].out + T - 1) / T, T, 0, stream>>>(yseg, bg[l],
                                                                         L[l].xout, n,
                                                                         L[l].out, S);
  }

  // --- final MLP + sigmoid ---
  final_mlp_kernel<<<(n + T - 1) / T, T, 0, stream>>>(xbuf1, Wf1, bf1, Wf2, bf2, Wf3, bf3,
                                                      (float*)d_out, n);
}